// RecommendationModel_58531814310253
// MI455X (gfx1250) — compile-verified
//
#include <hip/hip_runtime.h>
#include <math.h>

typedef __attribute__((ext_vector_type(2))) float v2f;
typedef __attribute__((ext_vector_type(8))) float v8f;
typedef __attribute__((ext_vector_type(4))) unsigned int v4u;
typedef __attribute__((ext_vector_type(8))) int v8i;
typedef __attribute__((ext_vector_type(4))) int v4i;

#define N_USER  100000
#define N_MOVIE 20000
#define NEDGE   2000000
#define NLABEL  1000000
#define HDIM    128
#define D_USER  24
#define D_MOVIE 18

static __device__ __forceinline__ v8f wmma_f32_k4(v2f a, v2f b, v8f c) {
  // D = A(16x4 f32) x B(4x16 f32) + C(16x16 f32) -> v_wmma_f32_16x16x4_f32
  return __builtin_amdgcn_wmma_f32_16x16x4_f32(false, a, false, b, (short)0, c, false, false);
}

// ---------------------------------------------------------------------------
__global__ void degree_kernel(const int* __restrict__ eu, const int* __restrict__ em,
                              float* __restrict__ cnt_u, float* __restrict__ cnt_m) {
  int e = blockIdx.x * blockDim.x + threadIdx.x;
  if (e < NEDGE) {
    unsafeAtomicAdd(&cnt_u[eu[e]], 1.0f);
    unsafeAtomicAdd(&cnt_m[em[e]], 1.0f);
  }
}

__global__ void invert_kernel(float* __restrict__ c, int n) {
  int i = blockIdx.x * blockDim.x + threadIdx.x;
  if (i < n) c[i] = 1.0f / fmaxf(c[i], 1.0f);
}

// scatter-add messages: agg[dst[e]] += xsrc[src[e]]   (one thread per (edge,feature))
template <int D>
__global__ void scatter_kernel(const float* __restrict__ xsrc, const int* __restrict__ src,
                               const int* __restrict__ dst, float* __restrict__ agg) {
  long long tid = (long long)blockIdx.x * blockDim.x + threadIdx.x;
  if (tid >= (long long)NEDGE * D) return;
  int e = (int)(tid / D);
  int f = (int)(tid % D);
  int s = src[e];
  int d = dst[e];
  unsafeAtomicAdd(&agg[(size_t)d * D + f], xsrc[(size_t)s * D + f]);
}

// ---------------------------------------------------------------------------
// Fused SAGE linear: out[M x 128] = (agg[m]*inv[m]) @ Wl + bl + X[m] @ Wr
// One wave per 16 rows, all 8 N-tiles accumulated in one K pass (A read once).
template <int KL, int KR>
__global__ void __launch_bounds__(32) sage_linear_kernel(
    const float* __restrict__ agg, const float* __restrict__ invc,
    const float* __restrict__ Wl,  const float* __restrict__ bl,
    const float* __restrict__ X,   const float* __restrict__ Wr,
    float* __restrict__ out) {
  const int lane = threadIdx.x;
  const int row  = lane & 15;   // A row / B,C,D column
  const int half = lane >> 4;
  const int m    = blockIdx.x * 16 + row;
  constexpr bool GL = (KL % 4) != 0;   // need K guards?
  constexpr bool GR = (KR % 4) != 0;

  v8f acc[8] = {};
  {
    const float ic = invc[m];
    const float* arow = agg + (size_t)m * KL;
    for (int k0 = 0; k0 < KL; k0 += 4) {
      const int kb = k0 + half * 2;
      v2f a;
      a.x = (!GL || kb     < KL) ? arow[kb]     * ic : 0.0f;
      a.y = (!GL || kb + 1 < KL) ? arow[kb + 1] * ic : 0.0f;
#pragma unroll
      for (int nt = 0; nt < 8; ++nt) {
        v2f b;
        b.x = (!GL || kb     < KL) ? Wl[(size_t)kb * HDIM + nt * 16 + row]       : 0.0f;
        b.y = (!GL || kb + 1 < KL) ? Wl[(size_t)(kb + 1) * HDIM + nt * 16 + row] : 0.0f;
        acc[nt] = wmma_f32_k4(a, b, acc[nt]);
      }
    }
  }
  {
    const float* xrow = X + (size_t)m * KR;
    for (int k0 = 0; k0 < KR; k0 += 4) {
      const int kb = k0 + half * 2;
      v2f a;
      a.x = (!GR || kb     < KR) ? xrow[kb]     : 0.0f;
      a.y = (!GR || kb + 1 < KR) ? xrow[kb + 1] : 0.0f;
#pragma unroll
      for (int nt = 0; nt < 8; ++nt) {
        v2f b;
        b.x = (!GR || kb     < KR) ? Wr[(size_t)kb * HDIM + nt * 16 + row]       : 0.0f;
        b.y = (!GR || kb + 1 < KR) ? Wr[(size_t)(kb + 1) * HDIM + nt * 16 + row] : 0.0f;
        acc[nt] = wmma_f32_k4(a, b, acc[nt]);
      }
    }
  }
#pragma unroll
  for (int nt = 0; nt < 8; ++nt) {
    const int n = nt * 16 + row;
    const float bn = bl[n];
#pragma unroll
    for (int r = 0; r < 8; ++r) {
      const int mm = blockIdx.x * 16 + r + 8 * half;   // D row layout
      out[(size_t)mm * HDIM + n] = acc[nt][r] + bn;
    }
  }
}

// ---------------------------------------------------------------------------
// Classifier head. 4 waves/block, 16 edges per wave (64 per block).
// wc1 (256x64 f32 = 64KB) staged into LDS via the Tensor Data Mover.
#define CW 4
#define H1S 66   // padded h1 row stride (floats): 8B-aligned, bank-conflict-free
#define H2S 17   // padded h2 row stride

__global__ void __launch_bounds__(32 * CW) classifier_kernel(
    const float* __restrict__ u2, const float* __restrict__ m2,
    const int* __restrict__ eu,  const int* __restrict__ em,
    const float* __restrict__ wc1, const float* __restrict__ bc1,
    const float* __restrict__ wc2, const float* __restrict__ bc2,
    const float* __restrict__ wc3, const float* __restrict__ bc3,
    float* __restrict__ out) {
  __shared__ float sw1[256 * 64];        // 64KB: wc1 staged by TDM
  __shared__ float h1[CW][16 * H1S];
  __shared__ float h2[CW][16 * H2S];

  // ---- TDM: DMA wc1 -> LDS (wave 0 issues; TENSORcnt tracks completion) ----
  if (threadIdx.x < 32) {
    const unsigned long long ga = (unsigned long long)(uintptr_t)wc1;
    const unsigned int lds = (unsigned int)(uintptr_t)&sw1[0];  // low 32b = LDS offset
    v4u g0;
    g0[0] = 1u;                                     // count=1 valid user D#
    g0[1] = lds;                                    // lds_addr
    g0[2] = (unsigned int)(ga & 0xffffffffu);       // global_addr[31:0]
    g0[3] = (unsigned int)((ga >> 32) & 0x01ffffffu) | (2u << 30);  // addr[56:32] | type=2
    v8i g1;
    g1[0] = 0x20000;                                // data_size=2 (4B), mask=0
    g1[1] = (int)(64u << 16);                       // tensor_dim0 = 64
    g1[2] = (int)(256u << 16);                      // tensor_dim1 = 256
    g1[3] = (int)(64u << 16);                       // tile_dim0 = 64
    g1[4] = 256;                                    // tile_dim1 = 256, tile_dim2 = 0
    g1[5] = 64;                                     // tensor_dim0_stride = 64
    g1[6] = 0;
    g1[7] = 0;
    v4i g2 = {0, 0, 0, 0};
    v4i g3 = {0, 0, 0, 0};
    v8i g4 = {0, 0, 0, 0, 0, 0, 0, 0};
    __builtin_amdgcn_tensor_load_to_lds(g0, g1, g2, g3, g4, 0);
    __builtin_amdgcn_s_wait_tensorcnt(0);
  }
  __syncthreads();

  const int wid  = threadIdx.x >> 5;
  const int lane = threadIdx.x & 31;
  const int row  = lane & 15;
  const int half = lane >> 4;
  const int e0   = (blockIdx.x * CW + wid) * 16;
  const int iu = eu[e0 + row];
  const int im = em[e0 + row];
  const float* rowU = u2 + (size_t)iu * HDIM;
  const float* rowM = m2 + (size_t)im * HDIM;
  float* h1w = h1[wid];
  float* h2w = h2[wid];

  // stage 1: feat(16x256) @ wc1(256x64); 4 N-tiles in one K pass
  v8f acc[4] = {};
  for (int k0 = 0; k0 < 256; k0 += 4) {
    const int kb = k0 + half * 2;                     // kb even -> no half-row split
    const float* rs = (kb < HDIM) ? (rowU + kb) : (rowM + (kb - HDIM));
    v2f a;
    a.x = rs[0];
    a.y = rs[1];
#pragma unroll
    for (int nt = 0; nt < 4; ++nt) {
      v2f b;
      b.x = sw1[kb * 64 + nt * 16 + row];
      b.y = sw1[(kb + 1) * 64 + nt * 16 + row];
      acc[nt] = wmma_f32_k4(a, b, acc[nt]);
    }
  }
#pragma unroll
  for (int nt = 0; nt < 4; ++nt) {
    const int n = nt * 16 + row;
    const float bb = bc1[n];
#pragma unroll
    for (int r = 0; r < 8; ++r)
      h1w[(r + 8 * half) * H1S + n] = fmaxf(acc[nt][r] + bb, 0.0f);
  }
  __syncthreads();

  // stage 2: h1(16x64) @ wc2(64x16) via LDS re-read in A layout
  v8f acc2 = {};
  for (int k0 = 0; k0 < 64; k0 += 4) {
    const int kb = k0 + half * 2;
    v2f a, b;
    a.x = h1w[row * H1S + kb];
    a.y = h1w[row * H1S + kb + 1];
    b.x = wc2[kb * 16 + row];
    b.y = wc2[(kb + 1) * 16 + row];
    acc2 = wmma_f32_k4(a, b, acc2);
  }
  {
    const float bb = bc2[row];
#pragma unroll
    for (int r = 0; r < 8; ++r)
      h2w[(r + 8 * half) * H2S + row] = fmaxf(acc2[r] + bb, 0.0f);
  }
  __syncthreads();

  // stage 3: dot with wc3, sigmoid * 5
  if (lane < 16) {
    float s = bc3[0];
#pragma unroll
    for (int n = 0; n < 16; ++n) s += h2w[lane * H2S + n] * wc3[n];
    out[e0 + lane] = 5.0f / (1.0f + __expf(-s));
  }
}

// ---------------------------------------------------------------------------
extern "C" void kernel_launch(void* const* d_in, const int* in_sizes, int n_in,
                              void* d_out, int out_size, void* d_ws, size_t ws_size,
                              hipStream_t stream) {
  const float* x_user      = (const float*)d_in[0];
  const float* x_movie     = (const float*)d_in[1];
  const int*   edge_user   = (const int*)d_in[2];
  const int*   edge_movie  = (const int*)d_in[3];
  const int*   elabel_user = (const int*)d_in[4];
  const int*   elabel_movie= (const int*)d_in[5];
  const float* w1m_l = (const float*)d_in[6];
  const float* b1m_l = (const float*)d_in[7];
  const float* w1m_r = (const float*)d_in[8];
  const float* w1u_l = (const float*)d_in[9];
  const float* b1u_l = (const float*)d_in[10];
  const float* w1u_r = (const float*)d_in[11];
  const float* w2m_l = (const float*)d_in[12];
  const float* b2m_l = (const float*)d_in[13];
  const float* w2m_r = (const float*)d_in[14];
  const float* w2u_l = (const float*)d_in[15];
  const float* b2u_l = (const float*)d_in[16];
  const float* w2u_r = (const float*)d_in[17];
  const float* wc1 = (const float*)d_in[18];
  const float* bc1 = (const float*)d_in[19];
  const float* wc2 = (const float*)d_in[20];
  const float* bc2 = (const float*)d_in[21];
  const float* wc3 = (const float*)d_in[22];
  const float* bc3 = (const float*)d_in[23];
  float* out = (float*)d_out;

  // workspace layout (floats), ~194 MB total
  float* ws = (float*)d_ws;
  size_t o = 0;
  float* cnt_user  = ws + o; o += N_USER;
  float* cnt_movie = ws + o; o += N_MOVIE;
  float* agg_mu = ws + o; o += (size_t)N_MOVIE * D_USER;
  float* agg_um = ws + o; o += (size_t)N_USER  * D_MOVIE;
  const size_t zero1_end = o;
  float* m1 = ws + o; o += (size_t)N_MOVIE * HDIM;
  float* u1 = ws + o; o += (size_t)N_USER  * HDIM;
  const size_t zero2_beg = o;
  float* agg_m2 = ws + o; o += (size_t)N_MOVIE * HDIM;
  float* agg_u2 = ws + o; o += (size_t)N_USER  * HDIM;
  const size_t zero2_end = o;
  float* m2 = ws + o; o += (size_t)N_MOVIE * HDIM;
  float* u2 = ws + o; o += (size_t)N_USER  * HDIM;

  (void)hipMemsetAsync(ws, 0, zero1_end * sizeof(float), stream);
  (void)hipMemsetAsync(ws + zero2_beg, 0, (zero2_end - zero2_beg) * sizeof(float), stream);

  degree_kernel<<<(NEDGE + 255) / 256, 256, 0, stream>>>(edge_user, edge_movie, cnt_user, cnt_movie);
  invert_kernel<<<(N_USER + N_MOVIE + 255) / 256, 256, 0, stream>>>(cnt_user, N_USER + N_MOVIE);

  // layer 1 aggregation
  scatter_kernel<D_USER><<<(int)(((long long)NEDGE * D_USER + 255) / 256), 256, 0, stream>>>(
      x_user, edge_user, edge_movie, agg_mu);
  scatter_kernel<D_MOVIE><<<(int)(((long long)NEDGE * D_MOVIE + 255) / 256), 256, 0, stream>>>(
      x_movie, edge_movie, edge_user, agg_um);

  // layer 1 linears (WMMA)
  sage_linear_kernel<D_USER, D_MOVIE><<<N_MOVIE / 16, 32, 0, stream>>>(
      agg_mu, cnt_movie, w1m_l, b1m_l, x_movie, w1m_r, m1);
  sage_linear_kernel<D_MOVIE, D_USER><<<N_USER / 16, 32, 0, stream>>>(
      agg_um, cnt_user, w1u_l, b1u_l, x_user, w1u_r, u1);

  // layer 2 aggregation (dominant memory cost, ~4 GB)
  scatter_kernel<HDIM><<<(int)(((long long)NEDGE * HDIM + 255) / 256), 256, 0, stream>>>(
      u1, edge_user, edge_movie, agg_m2);
  scatter_kernel<HDIM><<<(int)(((long long)NEDGE * HDIM + 255) / 256), 256, 0, stream>>>(
      m1, edge_movie, edge_user, agg_u2);

  // layer 2 linears (WMMA)
  sage_linear_kernel<HDIM, HDIM><<<N_MOVIE / 16, 32, 0, stream>>>(
      agg_m2, cnt_movie, w2m_l, b2m_l, m1, w2m_r, m2);
  sage_linear_kernel<HDIM, HDIM><<<N_USER / 16, 32, 0, stream>>>(
      agg_u2, cnt_user, w2u_l, b2u_l, u1, w2u_r, u2);

  // classifier head (WMMA + TDM), 64 edges per block
  classifier_kernel<<<NLABEL / (16 * CW), 32 * CW, 0, stream>>>(
      u2, m2, elabel_user, elabel_movie, wc1, bc1, wc2, bc2, wc3, bc3, out);
}